// SymmetricMaskedGraphConv_39565238731020
// MI455X (gfx1250) — compile-verified
//
#include <hip/hip_runtime.h>

// MI455X / gfx1250, wave32. f32 WMMA path (V_WMMA_F32_16X16X4_F32).
// Each wave computes a 16x64 output strip: one A fetch (and one sigmoid/exp
// in kernel 2) feeds 4 WMMAs per K-step.

typedef __attribute__((ext_vector_type(2))) float v2f;
typedef __attribute__((ext_vector_type(8))) float v8f;

#define NTOT   6400  // N
#define NNODE  200   // n, nodes per graph
#define NGRAPH 32    // B = N/n
#define FDIM   128   // F_in == F_out
#define KPAD   208   // 200 padded to multiple of 16
#define CTILES 4     // 16-col tiles per wave (16x64 strip)

// ---------------------------------------------------------------------------
// Kernel 1: support = feat @ weight   [6400,128] = [6400,128] x [128,128]
// One wave per 16x64 output strip; K stepped by 4 via f32 WMMA; A shared
// across the 4 column tiles.
// ---------------------------------------------------------------------------
__global__ __launch_bounds__(256) void support_gemm_kernel(
    const float* __restrict__ feat,     // [6400,128] row-major
    const float* __restrict__ weight,   // [128,128]  row-major
    float* __restrict__ support)        // [6400,128] row-major (workspace)
{
    const int lane = threadIdx.x & 31;
    const int wave = threadIdx.x >> 5;
    const int strip = blockIdx.x * 8 + wave;     // 400*2 = 800 strips
    const int CG = FDIM / (16 * CTILES);         // 2 column groups
    const int mt = strip / CG;                   // 0..399
    const int cg = strip % CG;                   // 0..1
    const int m0 = mt * 16;
    const int n0 = cg * 16 * CTILES;
    const int lc = lane & 15;
    const int h  = lane >> 4;

    const float* __restrict__ arow = feat + (m0 + lc) * FDIM;

    v8f acc[CTILES] = {};
    for (int k0 = 0; k0 < FDIM; k0 += 4) {
        // A 16x4: VGPR v, lane-half h -> K = v + 2h (contiguous pair per lane)
        v2f a;
        a.x = arow[k0 + 2 * h];
        a.y = arow[k0 + 2 * h + 1];
        // B 4x16 per column tile: K = v + 2h, N = lane%16
        const int krow = (k0 + 2 * h) * FDIM;
#pragma unroll
        for (int j = 0; j < CTILES; ++j) {
            const int c = n0 + 16 * j + lc;
            v2f b;
            b.x = weight[krow + c];
            b.y = weight[krow + FDIM + c];
            acc[j] = __builtin_amdgcn_wmma_f32_16x16x4_f32(
                false, a, false, b, (short)0, acc[j], false, false);
        }
    }

    // D 16x16 f32: VGPR r -> M = r + 8h, N = lane%16
#pragma unroll
    for (int j = 0; j < CTILES; ++j) {
#pragma unroll
        for (int r = 0; r < 8; ++r) {
            support[(m0 + r + 8 * h) * FDIM + n0 + 16 * j + lc] = acc[j][r];
        }
    }
}

// ---------------------------------------------------------------------------
// Kernel 2: per graph g:
//   out_g[200,128] = (sigmoid(adj_g) * mask) @ support_g
//   mask[i][j] = 0.5*(raw[i][j] + raw[j][i]) + (i==j)
// adj_g is the g-th diagonal 200x200 block of adj[6400,6400].
// M,K = 200 padded to 208; out-of-range A/B elements contribute 0.
// The masked-A element (incl. v_exp_f32 sigmoid) is computed ONCE per
// K-step and reused by 4 WMMAs.
// ---------------------------------------------------------------------------
__global__ __launch_bounds__(256) void masked_graph_gemm_kernel(
    const float* __restrict__ adj,      // [6400,6400]
    const float* __restrict__ raw,      // [200,200]
    const float* __restrict__ support,  // [6400,128] (= [32,200,128])
    float* __restrict__ out)            // [6400,128]
{
    const int lane = threadIdx.x & 31;
    const int wave = threadIdx.x >> 5;
    int strip = blockIdx.x * 8 + wave;           // 32*13*2 = 832 strips
    const int MT = KPAD / 16;                    // 13
    const int CG = FDIM / (16 * CTILES);         // 2
    const int g  = strip / (MT * CG);            // graph index 0..31
    int t = strip % (MT * CG);
    const int mt = t / CG;
    const int cg = t % CG;
    const int m0 = mt * 16;
    const int n0 = cg * 16 * CTILES;
    const int lc = lane & 15;
    const int h  = lane >> 4;

    const int r    = m0 + lc;                    // A row within graph, 0..207
    const bool rok = (r < NNODE);
    const int gbase = (g * NNODE) * NTOT + g * NNODE;  // adj block origin
    const int srow0 = g * NNODE;                 // support row base for graph

    v8f acc[CTILES] = {};
    for (int k0 = 0; k0 < KPAD; k0 += 4) {
        const int ka = k0 + 2 * h;               // even; pair is (ka, ka+1)
        const bool kok = (ka < NNODE);           // NNODE even => covers ka+1

        // A = sigmoid(adj) * mask, generated in-register once per K-step
        v2f a;
        a.x = 0.0f;
        a.y = 0.0f;
        if (rok && kok) {
            const float x0 = adj[gbase + r * NTOT + ka];
            const float x1 = adj[gbase + r * NTOT + ka + 1];
            const float s0 = 1.0f / (1.0f + __expf(-x0));
            const float s1 = 1.0f / (1.0f + __expf(-x1));
            const float m0v = 0.5f * (raw[r * NNODE + ka]     + raw[ka * NNODE + r])
                            + (r == ka     ? 1.0f : 0.0f);
            const float m1v = 0.5f * (raw[r * NNODE + ka + 1] + raw[(ka + 1) * NNODE + r])
                            + (r == ka + 1 ? 1.0f : 0.0f);
            a.x = s0 * m0v;
            a.y = s1 * m1v;
        }

        // B from support, 4 column tiles (padded K rows read as 0)
        const int brow = (srow0 + ka) * FDIM;
#pragma unroll
        for (int j = 0; j < CTILES; ++j) {
            const int c = n0 + 16 * j + lc;
            v2f b;
            b.x = kok ? support[brow + c]        : 0.0f;
            b.y = kok ? support[brow + FDIM + c] : 0.0f;
            acc[j] = __builtin_amdgcn_wmma_f32_16x16x4_f32(
                false, a, false, b, (short)0, acc[j], false, false);
        }
    }

    // Store rows < 200 only
#pragma unroll
    for (int j = 0; j < CTILES; ++j) {
#pragma unroll
        for (int rr = 0; rr < 8; ++rr) {
            const int row = m0 + rr + 8 * h;
            if (row < NNODE) {
                out[(g * NNODE + row) * FDIM + n0 + 16 * j + lc] = acc[j][rr];
            }
        }
    }
}

// ---------------------------------------------------------------------------
extern "C" void kernel_launch(void* const* d_in, const int* in_sizes, int n_in,
                              void* d_out, int out_size, void* d_ws, size_t ws_size,
                              hipStream_t stream) {
    const float* adj    = (const float*)d_in[0];   // [6400,6400]
    const float* feat   = (const float*)d_in[1];   // [6400,128]
    const float* weight = (const float*)d_in[2];   // [128,128]
    const float* raw    = (const float*)d_in[3];   // [200,200]
    float* out = (float*)d_out;                    // [6400,128]
    float* support = (float*)d_ws;                 // [6400,128] scratch (3.3 MB)

    // Kernel 1: 400 M-tiles * 2 column groups = 800 strips -> 100 blocks
    support_gemm_kernel<<<100, 256, 0, stream>>>(feat, weight, support);

    // Kernel 2: 32 graphs * 13 M-tiles * 2 column groups = 832 strips -> 104 blocks
    masked_graph_gemm_kernel<<<104, 256, 0, stream>>>(adj, raw, support, out);
}